// HAMA_82188494176606
// MI455X (gfx1250) — compile-verified
//
#include <hip/hip_runtime.h>
#include <hip/hip_bf16.h>
#include <math.h>

// ---------------- problem constants (fixed by the reference) ----------------
#define NTOT   16384      // B * N_NODE
#define NNODE  1024
#define BG     16         // graphs
#define CCLU   16         // clusters per graph
#define NC     (BG*CCLU)  // 256 cluster nodes
#define HID    128
#define HEADS  4
#define H4D    (HEADS*HID)   // 512
#define INF    32
#define NEG_SLOPE 0.2f

typedef __attribute__((ext_vector_type(2))) float v2f;
typedef __attribute__((ext_vector_type(8))) float v8f;

// =====================================================================
// fp32 WMMA GEMM with async-LDS B staging:
//   C[M,N] = A[M,K] @ B[K,N] (+bias) (+resid) (tanh)
// Wave tile 16(M) x 64(N); block = 8 waves (4 along M x 2 along N)
//   -> block tile 64(M) x 128(N).
// B panel is staged per block into LDS in 32-row K-chunks using
// GLOBAL_LOAD_ASYNC_TO_LDS_B128 (ASYNCcnt) -- each M-wave then reads the
// shared tile via ds_load instead of 4x-redundant global loads.
// REQUIRES: M%64==0, N%128==0, K%32==0 (true for every call below), so the
// grid covers exactly and every wave participates in the barriers.
// Fragment layouts per CDNA5 ISA (f32 16x16x4):
//   A: VGPR0 = K+0 (lanes 0-15) / K+2 (lanes 16-31); VGPR1 = K+1 / K+3
//   B: mirror: lanes 0-15 rows K+0/K+1, lanes 16-31 rows K+2/K+3, N = lane&15
//   C/D: VGPR j -> M = j (lanes 0-15), M = j+8 (lanes 16-31), N = lane&15
// =====================================================================
#define KC 32
__global__ __launch_bounds__(256)
void wmma_gemm_f32(const float* __restrict__ A, const float* __restrict__ B,
                   const float* __restrict__ bias, const float* __restrict__ resid,
                   float* __restrict__ C, int M, int N, int K, int act)
{
    __shared__ float Bs[KC * 128];               // 16 KB

    const int lane = threadIdx.x & 31;
    const int wave = threadIdx.x >> 5;
    const int row0 = blockIdx.y * 64 + (wave & 3) * 16;
    const int colB = blockIdx.x * 128;           // block's B/C column base
    const int col0 = colB + (wave >> 2) * 64;    // this wave's column base

    const int ar = row0 + (lane & 15);
    const int kk = (lane >> 4) * 2;              // 0 (lo lane half) or 2 (hi half)
    const int bn = (col0 - colB) + (lane & 15);  // column within the LDS tile

    // staging assignment: thread t copies 16 consecutive floats of one B row
    const int t  = threadIdx.x;
    const int kr = t >> 3;                       // 0..31 (chunk row)
    const int cb = (t & 7) << 4;                 // 0,16,...,112 (float col)
    const uint32_t lds_dst =
        (uint32_t)(uintptr_t)&Bs[kr * 128 + cb]; // low 32 bits of a generic LDS
                                                 // pointer == LDS byte offset

    v8f acc0 = {}, acc1 = {}, acc2 = {}, acc3 = {};

    for (int kc = 0; kc < K; kc += KC) {
        __syncthreads();                         // LDS reuse across chunks
        {
            // async DMA: 4 x b128 = 64 contiguous bytes per thread; the 24-bit
            // IOFFSET advances BOTH the LDS and global address (ISA 08 §4.4)
            uint64_t g = (uint64_t)(uintptr_t)(B + (size_t)(kc + kr) * N + colB + cb);
            asm volatile(
                "global_load_async_to_lds_b128 %0, %1, off\n\t"
                "global_load_async_to_lds_b128 %0, %1, off offset:16\n\t"
                "global_load_async_to_lds_b128 %0, %1, off offset:32\n\t"
                "global_load_async_to_lds_b128 %0, %1, off offset:48"
                :: "v"(lds_dst), "v"(g) : "memory");
            asm volatile("s_wait_asynccnt 0" ::: "memory");
        }
        __syncthreads();

        #pragma unroll
        for (int k2 = 0; k2 < KC; k2 += 4) {
            const int k = kc + k2;
            v2f a;
            a.x = A[(size_t)ar * K + k + kk];
            a.y = A[(size_t)ar * K + k + kk + 1];
            const float* Br0 = &Bs[(k2 + kk) * 128 + bn];
            const float* Br1 = Br0 + 128;
            v2f b0, b1, b2, b3;
            b0.x = Br0[0];  b0.y = Br1[0];
            b1.x = Br0[16]; b1.y = Br1[16];
            b2.x = Br0[32]; b2.y = Br1[32];
            b3.x = Br0[48]; b3.y = Br1[48];
            acc0 = __builtin_amdgcn_wmma_f32_16x16x4_f32(false, a, false, b0, (short)0, acc0, false, false);
            acc1 = __builtin_amdgcn_wmma_f32_16x16x4_f32(false, a, false, b1, (short)0, acc1, false, false);
            acc2 = __builtin_amdgcn_wmma_f32_16x16x4_f32(false, a, false, b2, (short)0, acc2, false, false);
            acc3 = __builtin_amdgcn_wmma_f32_16x16x4_f32(false, a, false, b3, (short)0, acc3, false, false);
        }
    }

#define GEMM_EPILOGUE(ACC, T)                                              \
    _Pragma("unroll")                                                      \
    for (int j = 0; j < 8; ++j) {                                          \
        int row = row0 + j + ((lane >= 16) ? 8 : 0);                       \
        int col = col0 + (T) * 16 + (lane & 15);                           \
        float v = ACC[j];                                                  \
        if (bias)  v += bias[col];                                         \
        if (act)   v = tanhf(v);                                           \
        if (resid) v += resid[(size_t)row * N + col];                      \
        C[(size_t)row * N + col] = v;                                      \
    }
    GEMM_EPILOGUE(acc0, 0)
    GEMM_EPILOGUE(acc1, 1)
    GEMM_EPILOGUE(acc2, 2)
    GEMM_EPILOGUE(acc3, 3)
#undef GEMM_EPILOGUE
}

// ---------------- ordered-int encoding for float atomic max ----------------
__device__ __forceinline__ int   f2ord(float f){ int i = __float_as_int(f); return i >= 0 ? i : (i ^ 0x7FFFFFFF); }
__device__ __forceinline__ float ord2f(int i){ return __int_as_float(i >= 0 ? i : (i ^ 0x7FFFFFFF)); }

// zero agg[n*128], z[n*4]; m[n*4] = -inf (ordered INT_MIN)
__global__ __launch_bounds__(256)
void init_level(float* __restrict__ agg, int* __restrict__ m, float* __restrict__ z, int n)
{
    int i = blockIdx.x * blockDim.x + threadIdx.x;
    if (i < n * HID) agg[i] = 0.0f;
    if (i < n * HEADS) { m[i] = (int)0x80000000; z[i] = 0.0f; }
}

__device__ __forceinline__ void edge_endpoints(int e, int mode,
                                               const int* __restrict__ esrc,
                                               const int* __restrict__ edst,
                                               int nbase, int& src, int& dst)
{
    if (mode == 0) {                       // low level: edge list + appended self loops
        if (e < nbase) { src = esrc[e]; dst = edst[e]; }
        else           { src = dst = e - nbase; }
    } else {                               // high level: fully connected per graph
        int b = e / (CCLU * CCLU);
        int r = e % (CCLU * CCLU);
        src = b * CCLU + r / CCLU;
        dst = b * CCLU + r % CCLU;
    }
}

// one wave per edge: logit[e,h] = sum_d lrelu(xl[src]+xr[dst]) * att[h,d]; atomic-max into m[dst,h]
__global__ __launch_bounds__(256)
void edge_logit_max(const float* __restrict__ xl, const float* __restrict__ xr,
                    const float* __restrict__ att,
                    const int* __restrict__ esrc, const int* __restrict__ edst,
                    int nbase, int ntot, int mode,
                    float* __restrict__ logit, int* __restrict__ m)
{
    int w = (blockIdx.x * blockDim.x + threadIdx.x) >> 5;
    int lane = threadIdx.x & 31;
    if (w >= ntot) return;
    int src, dst; edge_endpoints(w, mode, esrc, edst, nbase, src, dst);
    const float* xls = xl + (size_t)src * H4D;
    const float* xrd = xr + (size_t)dst * H4D;
    #pragma unroll
    for (int h = 0; h < HEADS; ++h) {
        float acc = 0.0f;
        #pragma unroll
        for (int i = 0; i < 4; ++i) {
            int d = lane + 32 * i;
            float v = xls[h * HID + d] + xrd[h * HID + d];
            v = v > 0.0f ? v : NEG_SLOPE * v;
            acc += v * att[h * HID + d];
        }
        #pragma unroll
        for (int off = 16; off > 0; off >>= 1) acc += __shfl_xor(acc, off, 32);
        if (lane == 0) {
            logit[(size_t)w * HEADS + h] = acc;
            atomicMax(&m[dst * HEADS + h], f2ord(acc));
        }
    }
}

// one thread per (edge, head): p = exp(logit - m[dst]); z[dst] += p (overwrite logit with p)
__global__ __launch_bounds__(256)
void edge_expsum(float* __restrict__ logit, const int* __restrict__ m, float* __restrict__ z,
                 const int* __restrict__ edst, int nbase, int ntot, int mode)
{
    int i = blockIdx.x * blockDim.x + threadIdx.x;
    if (i >= ntot * HEADS) return;
    int e = i >> 2, h = i & 3;
    int dst;
    if (mode == 0) dst = (e < nbase) ? edst[e] : (e - nbase);
    else           dst = (e / (CCLU * CCLU)) * CCLU + (e % CCLU);
    float p = __expf(logit[i] - ord2f(m[dst * HEADS + h]));
    logit[i] = p;
    atomicAdd(&z[dst * HEADS + h], p);
}

// one wave per edge: agg[dst,d] += (1/H) * sum_h (p/z[dst,h]) * xl[src,h,d]
__global__ __launch_bounds__(256)
void edge_aggregate(const float* __restrict__ p, const float* __restrict__ z,
                    const float* __restrict__ xl,
                    const int* __restrict__ esrc, const int* __restrict__ edst,
                    int nbase, int ntot, int mode, float* __restrict__ agg)
{
    int w = (blockIdx.x * blockDim.x + threadIdx.x) >> 5;
    int lane = threadIdx.x & 31;
    if (w >= ntot) return;
    int src, dst; edge_endpoints(w, mode, esrc, edst, nbase, src, dst);
    float alpha[HEADS];
    #pragma unroll
    for (int h = 0; h < HEADS; ++h)
        alpha[h] = p[(size_t)w * HEADS + h] / z[dst * HEADS + h];
    const float* xls = xl + (size_t)src * H4D;
    float* ad = agg + (size_t)dst * HID;
    #pragma unroll
    for (int i = 0; i < 4; ++i) {
        int d = lane + 32 * i;
        float v = 0.0f;
        #pragma unroll
        for (int h = 0; h < HEADS; ++h) v += alpha[h] * xls[h * HID + d];
        atomicAdd(&ad[d], v * (1.0f / HEADS));
    }
}

// one wave per node: h = LN(agg + bias)*g+beta [silu] + h (+hres2)
__global__ __launch_bounds__(256)
void node_update(const float* __restrict__ agg, const float* __restrict__ bias,
                 const float* __restrict__ g, const float* __restrict__ beta,
                 float* __restrict__ h, const float* __restrict__ hres2,
                 int do_silu, int n)
{
    int w = (blockIdx.x * blockDim.x + threadIdx.x) >> 5;
    int lane = threadIdx.x & 31;
    if (w >= n) return;
    float x[4], lsum = 0.0f;
    #pragma unroll
    for (int i = 0; i < 4; ++i) {
        x[i] = agg[(size_t)w * HID + lane + 32 * i] + bias[lane + 32 * i];
        lsum += x[i];
    }
    #pragma unroll
    for (int off = 16; off > 0; off >>= 1) lsum += __shfl_xor(lsum, off, 32);
    float mu = lsum * (1.0f / HID);
    float lvar = 0.0f;
    #pragma unroll
    for (int i = 0; i < 4; ++i) { float tt = x[i] - mu; lvar += tt * tt; }
    #pragma unroll
    for (int off = 16; off > 0; off >>= 1) lvar += __shfl_xor(lvar, off, 32);
    float rstd = rsqrtf(lvar * (1.0f / HID) + 1e-5f);
    #pragma unroll
    for (int i = 0; i < 4; ++i) {
        int d = lane + 32 * i;
        float y = (x[i] - mu) * rstd * g[d] + beta[d];
        if (do_silu) y = y / (1.0f + __expf(-y));
        float v = y + h[(size_t)w * HID + d];
        if (hres2) v += hres2[(size_t)w * HID + d];
        h[(size_t)w * HID + d] = v;
    }
}

// one thread per node: s = softmax(h @ pool_W + pool_b) over 16 clusters
__global__ __launch_bounds__(256)
void pool_softmax(const float* __restrict__ h, const float* __restrict__ W,
                  const float* __restrict__ b, float* __restrict__ s, int n)
{
    int i = blockIdx.x * blockDim.x + threadIdx.x;
    if (i >= n) return;
    float acc[CCLU];
    #pragma unroll
    for (int c = 0; c < CCLU; ++c) acc[c] = b[c];
    for (int k = 0; k < HID; ++k) {
        float hv = h[(size_t)i * HID + k];
        #pragma unroll
        for (int c = 0; c < CCLU; ++c) acc[c] += hv * W[k * CCLU + c];
    }
    float mx = acc[0];
    #pragma unroll
    for (int c = 1; c < CCLU; ++c) mx = fmaxf(mx, acc[c]);
    float ssum = 0.0f;
    #pragma unroll
    for (int c = 0; c < CCLU; ++c) { acc[c] = __expf(acc[c] - mx); ssum += acc[c]; }
    float inv = 1.0f / ssum;
    #pragma unroll
    for (int c = 0; c < CCLU; ++c) s[(size_t)i * CCLU + c] = acc[c] * inv;
}

// block per (c,b), 128 threads (d): Hf[b,c,d] = sum_n s[b,n,c]*h[b,n,d] / max(sum_n s, 1e-5)
__global__ __launch_bounds__(128)
void pool_reduce(const float* __restrict__ s, const float* __restrict__ h, float* __restrict__ Hf)
{
    int c = blockIdx.x, b = blockIdx.y, d = threadIdx.x;
    float acc = 0.0f, cnt = 0.0f;
    for (int nn = 0; nn < NNODE; ++nn) {
        size_t node = (size_t)b * NNODE + nn;
        float sv = s[node * CCLU + c];             // broadcast load within block
        acc += sv * h[node * HID + d];
        cnt += sv;
    }
    cnt = fmaxf(cnt, 1e-5f);
    Hf[((size_t)b * CCLU + c) * HID + d] = acc / cnt;
}

// one thread per (node, d): X = s @ Hf(per graph) + h0
__global__ __launch_bounds__(256)
void unpool_add(const float* __restrict__ s, const float* __restrict__ Hc,
                const float* __restrict__ h0, float* __restrict__ X)
{
    int i = blockIdx.x * blockDim.x + threadIdx.x;
    if (i >= NTOT * HID) return;
    int nidx = i >> 7, d = i & 127;
    int b = nidx >> 10;
    float acc = h0[i];
    #pragma unroll
    for (int c = 0; c < CCLU; ++c)
        acc += s[(size_t)nidx * CCLU + c] * Hc[((size_t)b * CCLU + c) * HID + d];
    X[i] = acc;
}

// =====================================================================
static inline void gemm(const float* A, const float* B, const float* bias, const float* resid,
                        float* C, int M, int N, int K, int act, hipStream_t st)
{
    dim3 grid((N + 127) / 128, (M + 63) / 64);
    wmma_gemm_f32<<<grid, 256, 0, st>>>(A, B, bias, resid, C, M, N, K, act);
}

extern "C" void kernel_launch(void* const* d_in, const int* in_sizes, int n_in,
                              void* d_out, int out_size, void* d_ws, size_t ws_size,
                              hipStream_t stream)
{
    const float* obs    = (const float*)d_in[0];
    const int*   eidx   = (const int*)d_in[1];
    const int    Eb     = in_sizes[1] / 2;              // 131072 base edges
    const float* emb_W  = (const float*)d_in[3];
    const float* emb_b  = (const float*)d_in[4];
    const float* pool_W = (const float*)d_in[5];
    const float* pool_b = (const float*)d_in[6];
    const float* mlp_W1 = (const float*)d_in[7];
    const float* mlp_b1 = (const float*)d_in[8];
    const float* mlp_W2 = (const float*)d_in[9];
    const float* mlp_b2 = (const float*)d_in[10];
    const float* gatp[2][8];                             // low=0, high=1: Wl,bl,Wr,br,att,bias,g,beta
    for (int lv = 0; lv < 2; ++lv)
        for (int t = 0; t < 8; ++t)
            gatp[lv][t] = (const float*)d_in[11 + lv * 8 + t];
    float* out = (float*)d_out;

    // ---- carve workspace (everything L2-resident: ~104 MB total) ----
    float* p   = (float*)d_ws;
    float* h    = p; p += (size_t)NTOT * HID;
    float* h0   = p; p += (size_t)NTOT * HID;
    float* X    = p; p += (size_t)NTOT * HID;
    float* agg  = p; p += (size_t)NTOT * HID;
    float* xl   = p; p += (size_t)NTOT * H4D;
    float* xr   = p; p += (size_t)NTOT * H4D;
    float* logit= p; p += (size_t)(Eb + NTOT) * HEADS;
    int*   mI   = (int*)p; p += (size_t)NTOT * HEADS;
    float* z    = p; p += (size_t)NTOT * HEADS;
    float* s    = p; p += (size_t)NTOT * CCLU;
    float* hc   = p; p += (size_t)NC * HID;
    float* hcres= p; p += (size_t)NC * HID;

    const int ELlow = Eb + NTOT;                         // + self loops
    const int ELhi  = BG * CCLU * CCLU;

    // 1) embedding: h0 = obs @ emb_W + emb_b ; h = h0
    gemm(obs, emb_W, emb_b, nullptr, h0, NTOT, HID, INF, 0, stream);
    hipMemcpyAsync(h, h0, (size_t)NTOT * HID * sizeof(float), hipMemcpyDeviceToDevice, stream);

    // 2) low-level GAT stack (stack residual == h0)
    for (int j = 0; j < 2; ++j) {
        gemm(h, gatp[0][0] + (size_t)j * HID * H4D, gatp[0][1] + j * H4D, nullptr, xl, NTOT, H4D, HID, 0, stream);
        gemm(h, gatp[0][2] + (size_t)j * HID * H4D, gatp[0][3] + j * H4D, nullptr, xr, NTOT, H4D, HID, 0, stream);
        init_level<<<(NTOT * HID + 255) / 256, 256, 0, stream>>>(agg, mI, z, NTOT);
        edge_logit_max<<<(ELlow + 7) / 8, 256, 0, stream>>>(xl, xr, gatp[0][4] + j * HEADS * HID,
                                                            eidx, eidx + Eb, Eb, ELlow, 0, logit, mI);
        edge_expsum<<<(ELlow * HEADS + 255) / 256, 256, 0, stream>>>(logit, mI, z, eidx + Eb, Eb, ELlow, 0);
        edge_aggregate<<<(ELlow + 7) / 8, 256, 0, stream>>>(logit, z, xl, eidx, eidx + Eb, Eb, ELlow, 0, agg);
        node_update<<<(NTOT + 7) / 8, 256, 0, stream>>>(agg, gatp[0][5] + j * HID, gatp[0][6] + j * HID,
                                                        gatp[0][7] + j * HID, h,
                                                        (j == 1) ? h0 : nullptr, (j == 0) ? 1 : 0, NTOT);
    }

    // 3) soft pooling
    pool_softmax<<<(NTOT + 255) / 256, 256, 0, stream>>>(h, pool_W, pool_b, s, NTOT);
    pool_reduce<<<dim3(CCLU, BG), 128, 0, stream>>>(s, h, hc);
    hipMemcpyAsync(hcres, hc, (size_t)NC * HID * sizeof(float), hipMemcpyDeviceToDevice, stream);

    // 4) high-level GAT stack on 256 cluster nodes (analytic fully-connected edges)
    for (int j = 0; j < 2; ++j) {
        gemm(hc, gatp[1][0] + (size_t)j * HID * H4D, gatp[1][1] + j * H4D, nullptr, xl, NC, H4D, HID, 0, stream);
        gemm(hc, gatp[1][2] + (size_t)j * HID * H4D, gatp[1][3] + j * H4D, nullptr, xr, NC, H4D, HID, 0, stream);
        init_level<<<(NC * HID + 255) / 256, 256, 0, stream>>>(agg, mI, z, NC);
        edge_logit_max<<<(ELhi + 7) / 8, 256, 0, stream>>>(xl, xr, gatp[1][4] + j * HEADS * HID,
                                                           nullptr, nullptr, 0, ELhi, 1, logit, mI);
        edge_expsum<<<(ELhi * HEADS + 255) / 256, 256, 0, stream>>>(logit, mI, z, nullptr, 0, ELhi, 1);
        edge_aggregate<<<(ELhi + 7) / 8, 256, 0, stream>>>(logit, z, xl, nullptr, nullptr, 0, ELhi, 1, agg);
        node_update<<<(NC + 7) / 8, 256, 0, stream>>>(agg, gatp[1][5] + j * HID, gatp[1][6] + j * HID,
                                                      gatp[1][7] + j * HID, hc,
                                                      (j == 1) ? hcres : nullptr, (j == 0) ? 1 : 0, NC);
    }

    // 5) unpool + residual MLP: out = tanh(X@W1+b1)@W2 + b2 + X
    unpool_add<<<(NTOT * HID + 255) / 256, 256, 0, stream>>>(s, hc, h0, X);
    gemm(X, mlp_W1, mlp_b1, nullptr, xl /*reuse as [N,512] hidden*/, NTOT, 4 * HID, HID, 1, stream);
    gemm(xl, mlp_W2, mlp_b2, X, out, NTOT, HID, 4 * HID, 0, stream);
}